// RecurrentGCN_19104014533429
// MI455X (gfx1250) — compile-verified
//
#include <hip/hip_runtime.h>
#include <hip/hip_bf16.h>

typedef __attribute__((ext_vector_type(2))) float v2f;
typedef __attribute__((ext_vector_type(8))) float v8f;

#define F_DIM 35
#define CIN 70
#define KPAD 72
#define NPAD 48
#define KSTEPS 18
#define ROWS_PER_WAVE 16
#define WAVES_PER_BLOCK 8
#define ROWS_PER_BLOCK 128

#if __has_builtin(__builtin_amdgcn_wmma_f32_16x16x4_f32)
#define HAVE_WMMA_F32 1
#endif

// ---------------- utility / pointwise kernels ----------------

__global__ void zero_f(float* __restrict__ p, long cnt) {
  long i = (long)blockIdx.x * blockDim.x + threadIdx.x;
  if (i < cnt) p[i] = 0.f;
}

__global__ void degrees_k(const int* __restrict__ src, const int* __restrict__ dst,
                          const float* __restrict__ w, float* __restrict__ dout,
                          float* __restrict__ din, int E) {
  int e = blockIdx.x * blockDim.x + threadIdx.x;
  if (e < E) {
    float we = w[e];
    atomicAdd(&dout[src[e]], we);
    atomicAdd(&din[dst[e]], we);
  }
}

__global__ void normw_k(const int* __restrict__ src, const int* __restrict__ dst,
                        const float* __restrict__ w, const float* __restrict__ dout,
                        const float* __restrict__ din, float* __restrict__ wf,
                        float* __restrict__ wb, int E) {
  int e = blockIdx.x * blockDim.x + threadIdx.x;
  if (e < E) {
    float we = w[e];
    float d0 = dout[src[e]];
    float d1 = din[dst[e]];
    wf[e] = we / (d0 > 0.f ? d0 : 1.f);
    wb[e] = we / (d1 > 0.f ? d1 : 1.f);
  }
}

__global__ void build_xh_k(const float* __restrict__ x, const float* __restrict__ h,
                           float* __restrict__ xh, int n) {
  long i = (long)blockIdx.x * blockDim.x + threadIdx.x;
  long tot = (long)n * F_DIM;
  if (i < tot) {
    long row = i / F_DIM;
    int c = (int)(i % F_DIM);
    xh[row * CIN + c] = x[i];
    xh[row * CIN + F_DIM + c] = h[i];
  }
}

// one thread per (edge, feature-pair): Xout[to[e]] += w[e] * Xin[from[e]]
__global__ void diffuse_k(const float* __restrict__ Xin, float* __restrict__ Xout,
                          const int* __restrict__ from, const int* __restrict__ to,
                          const float* __restrict__ w, int E) {
  long i = (long)blockIdx.x * blockDim.x + threadIdx.x;
  long tot = (long)E * F_DIM;  // 35 float2's per edge == 70 floats
  if (i >= tot) return;
  int e = (int)(i / F_DIM);
  int f = (int)(i % F_DIM) * 2;
  float ww = w[e];
  long ain = (long)from[e] * CIN + f;
  long aout = (long)to[e] * CIN + f;
  atomicAdd(&Xout[aout], ww * Xin[ain]);
  atomicAdd(&Xout[aout + 1], ww * Xin[ain + 1]);
}

__global__ void sigmoid_k(const float* __restrict__ g, float* __restrict__ z, long cnt) {
  long i = (long)blockIdx.x * blockDim.x + threadIdx.x;
  if (i < cnt) z[i] = 1.f / (1.f + __expf(-g[i]));
}

// xh[:,35:70] = sigmoid(g) * h   (x half of xh untouched)
__global__ void applyr_k(const float* __restrict__ g, const float* __restrict__ h,
                         float* __restrict__ xh, int n) {
  long i = (long)blockIdx.x * blockDim.x + threadIdx.x;
  long tot = (long)n * F_DIM;
  if (i < tot) {
    long row = i / F_DIM;
    int c = (int)(i % F_DIM);
    float r = 1.f / (1.f + __expf(-g[i]));
    xh[row * CIN + F_DIM + c] = r * h[i];
  }
}

// h = z*h + (1-z)*tanh(g)
__global__ void updateh_k(const float* __restrict__ g, const float* __restrict__ z,
                          float* __restrict__ h, long cnt) {
  long i = (long)blockIdx.x * blockDim.x + threadIdx.x;
  if (i < cnt) {
    float c = tanhf(g[i]);
    float zz = z[i];
    h[i] = zz * h[i] + (1.f - zz) * c;
  }
}

__global__ void finalout_k(const float* __restrict__ h, const float* __restrict__ Wl,
                           const float* __restrict__ bl, float* __restrict__ out, int n) {
  int i = blockIdx.x * blockDim.x + threadIdx.x;
  if (i < n) {
    float s = bl[0];
#pragma unroll
    for (int f = 0; f < F_DIM; ++f) s += h[(long)i * F_DIM + f] * Wl[f];
    out[i] = fmaxf(s, 0.f);
  }
}

// ---------------- WMMA GEMM: G {=, +=} X[n,70] @ Wpad[70,35] (+bias) ----------------
// fp32 WMMA (16x16x4) keeps full fp32 accuracy; GEMM flops are negligible vs the
// L2-resident scatter traffic, so there is no roofline reason to downcast.

__global__ __launch_bounds__(256) void dconv_gemm_wmma(
    const float* __restrict__ X, const float* __restrict__ Wa,
    const float* __restrict__ Wb, const float* __restrict__ bias,
    float* __restrict__ G, int n, int initAcc) {
  __shared__ float wsm[KPAD * NPAD];                       // padded weights, zero-filled
  __shared__ float xs[WAVES_PER_BLOCK][ROWS_PER_WAVE * KPAD];

  // cooperative weight stage (Wa [+ Wb] for the fused k=0 term)
  for (int i = threadIdx.x; i < KPAD * NPAD; i += blockDim.x) {
    int k = i / NPAD, c = i % NPAD;
    float v = 0.f;
    if (k < CIN && c < F_DIM) {
      v = Wa[k * F_DIM + c];
      if (Wb != nullptr) v += Wb[k * F_DIM + c];
    }
    wsm[i] = v;
  }

  const int wave = threadIdx.x >> 5;
  const int lane = threadIdx.x & 31;
  const int row0 = blockIdx.x * ROWS_PER_BLOCK + wave * ROWS_PER_WAVE;
  float* xw = xs[wave];

  // per-wave X tile stage (16 rows x 72 padded cols)
  for (int i = lane; i < ROWS_PER_WAVE * KPAD; i += 32) {
    int r = i / KPAD, c = i % KPAD;
    int row = row0 + r;
    if (row >= n) row = n - 1;  // clamp; stores are masked later
    xw[i] = (c < CIN) ? X[(long)row * CIN + c] : 0.f;
  }
  __syncthreads();

  const int m = lane & 15;
  const int khalf = lane >> 4;

#ifdef HAVE_WMMA_F32
  v8f acc[3];
  for (int nt = 0; nt < 3; ++nt) {
    int col = nt * 16 + m;
    if (initAcc) {
      float bv = (bias != nullptr && col < F_DIM) ? bias[col] : 0.f;
#pragma unroll
      for (int r = 0; r < 8; ++r) acc[nt][r] = bv;
    } else {
#pragma unroll
      for (int r = 0; r < 8; ++r) {
        int row = row0 + r + khalf * 8;
        if (row >= n) row = n - 1;
        acc[nt][r] = (col < F_DIM) ? G[(long)row * F_DIM + col] : 0.f;
      }
    }
  }

  // K loop: A frag (16x4 f32): lane m holds K = ks*4 + khalf*2 + {0,1}
  //         B frag (4x16 f32): same K pair, col = m
#pragma unroll
  for (int ks = 0; ks < KSTEPS; ++ks) {
    const int kk = ks * 4 + khalf * 2;
    v2f a;
    a.x = xw[m * KPAD + kk];
    a.y = xw[m * KPAD + kk + 1];
#pragma unroll
    for (int nt = 0; nt < 3; ++nt) {
      v2f b;
      b.x = wsm[kk * NPAD + nt * 16 + m];
      b.y = wsm[(kk + 1) * NPAD + nt * 16 + m];
      acc[nt] = __builtin_amdgcn_wmma_f32_16x16x4_f32(
          false, a, false, b, (short)0, acc[nt], false, false);
    }
  }

  // D layout: VGPR r -> row (r + khalf*8), col = m (+ 16*nt)
  for (int nt = 0; nt < 3; ++nt) {
    int col = nt * 16 + m;
    if (col < F_DIM) {
#pragma unroll
      for (int r = 0; r < 8; ++r) {
        int row = row0 + r + khalf * 8;
        if (row < n) G[(long)row * F_DIM + col] = acc[nt][r];
      }
    }
  }
#else
  // scalar fallback (keeps the build alive if the builtin is missing)
  if (lane < F_DIM) {
    for (int r = 0; r < ROWS_PER_WAVE; ++r) {
      int row = row0 + r;
      if (row >= n) continue;
      float s;
      if (initAcc) s = (bias != nullptr) ? bias[lane] : 0.f;
      else s = G[(long)row * F_DIM + lane];
      for (int k = 0; k < CIN; ++k) s += xw[r * KPAD + k] * wsm[k * NPAD + lane];
      G[(long)row * F_DIM + lane] = s;
    }
  }
#endif
}

// ---------------- host orchestration ----------------

extern "C" void kernel_launch(void* const* d_in, const int* in_sizes, int n_in,
                              void* d_out, int out_size, void* d_ws, size_t ws_size,
                              hipStream_t stream) {
  (void)n_in; (void)out_size; (void)ws_size;
  const float* x = (const float*)d_in[0];
  const int* ei = (const int*)d_in[1];
  const float* ew = (const float*)d_in[2];
  const float* thz = (const float*)d_in[3];
  const float* bz = (const float*)d_in[4];
  const float* thr = (const float*)d_in[5];
  const float* br = (const float*)d_in[6];
  const float* thc = (const float*)d_in[7];
  const float* bc = (const float*)d_in[8];
  const float* Wl = (const float*)d_in[9];
  const float* bl = (const float*)d_in[10];

  const int n = in_sizes[0] / F_DIM;
  const int E = in_sizes[2];
  const int* src = ei;
  const int* dst = ei + E;

  float* out = (float*)d_out;   // [n] relu output
  float* wf = out + n;          // A[0] = wf  (part of the returned tuple)
  float* wb = wf + E;           // A[1] = wb

  float* p = (float*)d_ws;
  float* deg_out = p; p += n;
  float* deg_in  = p; p += n;
  float* h  = p; p += (size_t)n * F_DIM;
  float* xh = p; p += (size_t)n * CIN;
  float* D0 = p; p += (size_t)n * CIN;
  float* D1 = p; p += (size_t)n * CIN;
  float* g  = p; p += (size_t)n * F_DIM;
  float* z  = p; p += (size_t)n * F_DIM;

  auto nblk = [](long total, int tpb) { return (unsigned)((total + tpb - 1) / tpb); };
  auto zero = [&](float* q, long cnt) {
    zero_f<<<nblk(cnt, 256), 256, 0, stream>>>(q, cnt);
  };
  auto gemm = [&](const float* Xm, const float* Wa, const float* Wb2,
                  const float* bias, int initAcc) {
    dconv_gemm_wmma<<<nblk(n, ROWS_PER_BLOCK), 256, 0, stream>>>(
        Xm, Wa, Wb2, bias, g, n, initAcc);
  };
  auto diffuse = [&](const float* Xin, float* Xout, const int* from, const int* to,
                     const float* w) {
    diffuse_k<<<nblk((long)E * F_DIM, 256), 256, 0, stream>>>(Xin, Xout, from, to, w, E);
  };

  const long TH_K = (long)CIN * F_DIM;  // 70*35 per (dir,hop) slice
  const long TH_D = 3 * TH_K;           // theta[1] offset (K_HOPS=3)

  auto dconv = [&](const float* xcat, const float* th, const float* bias) {
    // k=0, both directions fused: g = xcat @ (th[0,0]+th[1,0]) + bias
    gemm(xcat, th, th + TH_D, bias, 1);
    // forward hops
    zero(D0, (long)n * CIN); diffuse(xcat, D0, src, dst, wf);
    gemm(D0, th + TH_K, nullptr, nullptr, 0);
    zero(D1, (long)n * CIN); diffuse(D0, D1, src, dst, wf);
    gemm(D1, th + 2 * TH_K, nullptr, nullptr, 0);
    // backward hops
    zero(D0, (long)n * CIN); diffuse(xcat, D0, dst, src, wb);
    gemm(D0, th + TH_D + TH_K, nullptr, nullptr, 0);
    zero(D1, (long)n * CIN); diffuse(D0, D1, dst, src, wb);
    gemm(D1, th + TH_D + 2 * TH_K, nullptr, nullptr, 0);
  };

  // ---- adjacency normalization (also produces output A) ----
  zero(deg_out, n);
  zero(deg_in, n);
  degrees_k<<<nblk(E, 256), 256, 0, stream>>>(src, dst, ew, deg_out, deg_in, E);
  normw_k<<<nblk(E, 256), 256, 0, stream>>>(src, dst, ew, deg_out, deg_in, wf, wb, E);

  // ---- two GRU cells ----
  zero(h, (long)n * F_DIM);
  for (int cell = 0; cell < 2; ++cell) {
    build_xh_k<<<nblk((long)n * F_DIM, 256), 256, 0, stream>>>(x, h, xh, n);
    dconv(xh, thz, bz);
    sigmoid_k<<<nblk((long)n * F_DIM, 256), 256, 0, stream>>>(g, z, (long)n * F_DIM);
    dconv(xh, thr, br);
    applyr_k<<<nblk((long)n * F_DIM, 256), 256, 0, stream>>>(g, h, xh, n);
    dconv(xh, thc, bc);  // xh is now concat(x, r*h)
    updateh_k<<<nblk((long)n * F_DIM, 256), 256, 0, stream>>>(g, z, h, (long)n * F_DIM);
  }

  finalout_k<<<nblk(n, 256), 256, 0, stream>>>(h, Wl, bl, out, n);
}